// myLoss_43198781063802
// MI455X (gfx1250) — compile-verified
//
#include <hip/hip_runtime.h>

// CDNA5 (gfx1250) wave32 WMMA fragment types
typedef __attribute__((ext_vector_type(2))) float v2f;   // 16x4 f32 A / 4x16 f32 B: 2 VGPRs/lane
typedef __attribute__((ext_vector_type(8))) float v8f;   // 16x16 f32 C/D: 8 VGPRs/lane

#define WAVES_PER_BLOCK  8
#define BLOCKS_PER_BATCH 32
#define UNROLL           4   // tiles (of 64 points) per loop trip per wave

// Stage 1: each wave streams groups of 4x64-point tiles of the w-channel, forms
// squared differences, and accumulates with V_WMMA_F32_16X16X4_F32
// (A = d^2 tile, B = all-ones, C chained as the f32 accumulator).
__global__ __launch_bounds__(256)
void myLoss_stage1(const float* __restrict__ out_pc,
                   const float* __restrict__ gt_pc,
                   float* __restrict__ partials,
                   int N, int groups) {
    const int batch = blockIdx.x;
    const int tid   = threadIdx.x;
    const int lane  = tid & 31;
    const int wave  = tid >> 5;

    // a_i = out_pc[i, 3]
    const float a = out_pc[batch * 4 + 3];

    // ISA 16x4 f32 A layout: lanes 0-15 -> M=lane, K={0,1}; lanes 16-31 -> M=lane-16, K={2,3}
    const int m  = lane & 15;
    const int kb = (lane >> 4) * 2;
    const int laneOff = m * 4 + kb;   // point offset of this lane's first element in a tile

    const size_t base = (size_t)batch * (size_t)N;  // point index base for this batch
    const int wavesPerBatch = WAVES_PER_BLOCK * BLOCKS_PER_BATCH;
    const int wv = blockIdx.y * WAVES_PER_BLOCK + wave;

    v8f c = {0.f, 0.f, 0.f, 0.f, 0.f, 0.f, 0.f, 0.f};
    v2f ones;  ones.x = 1.0f;  ones.y = 1.0f;   // all-ones B: layout-independent

    for (int g = wv; g < groups; g += wavesPerBatch) {
        const int tbase = g * UNROLL;

        // Issue all 8 non-temporal loads first (one clause, one wait -> 8-deep MLP).
        float x[2 * UNROLL];
        int   j[2 * UNROLL];
#pragma unroll
        for (int u = 0; u < UNROLL; ++u) {
            const int j0 = (tbase + u) * 64 + laneOff;
            const int j1 = j0 + 1;
            j[2 * u]     = j0;
            j[2 * u + 1] = j1;
            const int jc0 = (j0 < N) ? j0 : (N - 1);   // clamp: address stays in-bounds
            const int jc1 = (j1 < N) ? j1 : (N - 1);
            // w channel of point j lives at ((base+j)*4 + 3); stream-once -> non-temporal
            x[2 * u]     = __builtin_nontemporal_load(gt_pc + ((base + (size_t)jc0) * 4 + 3));
            x[2 * u + 1] = __builtin_nontemporal_load(gt_pc + ((base + (size_t)jc1) * 4 + 3));
        }

        // 4 chained WMMAs: D = A x Ones + C accumulates the per-row sums of d^2.
        // EXEC is all-ones here (loop condition is wave-uniform; masking is branchless).
#pragma unroll
        for (int u = 0; u < UNROLL; ++u) {
            const float d0 = (j[2 * u]     < N) ? (a - x[2 * u])     : 0.0f;
            const float d1 = (j[2 * u + 1] < N) ? (a - x[2 * u + 1]) : 0.0f;
            v2f afrag;
            afrag.x = d0 * d0;
            afrag.y = d1 * d1;
            c = __builtin_amdgcn_wmma_f32_16x16x4_f32(
                    /*neg_a=*/false, afrag,
                    /*neg_b=*/false, ones,
                    /*c_mod=*/(short)0, c,
                    /*reuse_a=*/false, /*reuse_b=*/false);
        }
    }

    // C/D layout: lanes 0-15: VGPR r <-> M=r; lanes 16-31: VGPR r <-> M=r+8.
    // All columns are identical (B was all-ones), so tile total = col0 of lane0 + lane16.
    float s = c[0] + c[1] + c[2] + c[3] + c[4] + c[5] + c[6] + c[7];
    float wave_sum = __shfl(s, 0, 32) + __shfl(s, 16, 32);

    __shared__ float lds[WAVES_PER_BLOCK];
    if (lane == 0) lds[wave] = wave_sum;
    __syncthreads();
    if (tid == 0) {
        float bs = 0.0f;
        for (int w = 0; w < WAVES_PER_BLOCK; ++w) bs += lds[w];
        partials[(size_t)blockIdx.x * gridDim.y + blockIdx.y] = bs;  // deterministic, no atomics
    }
}

// Stage 2: reduce the 2048 block partials in double, scale by 1/B.
__global__ __launch_bounds__(256)
void myLoss_stage2(const float* __restrict__ partials,
                   float* __restrict__ out, int P, double invB) {
    __shared__ double lds[256];
    double acc = 0.0;
    for (int p = threadIdx.x; p < P; p += 256)
        acc += (double)partials[p];
    lds[threadIdx.x] = acc;
    __syncthreads();
    for (int off = 128; off > 0; off >>= 1) {
        if ((int)threadIdx.x < off) lds[threadIdx.x] += lds[threadIdx.x + off];
        __syncthreads();
    }
    if (threadIdx.x == 0)
        out[0] = (float)(lds[0] * invB);
}

extern "C" void kernel_launch(void* const* d_in, const int* in_sizes, int n_in,
                              void* d_out, int out_size, void* d_ws, size_t ws_size,
                              hipStream_t stream) {
    const float* out_pc = (const float*)d_in[0];   // [B, 4] f32
    const float* gt_pc  = (const float*)d_in[1];   // [B, N, 4] f32
    float*       out    = (float*)d_out;           // scalar f32
    float*       parts  = (float*)d_ws;            // B * BLOCKS_PER_BATCH floats (8 KB)

    const int B = in_sizes[0] / 4;                 // 64
    const int N = in_sizes[1] / (B * 4);           // 500000
    const int tiles  = (N + 63) / 64;              // 64-point WMMA tiles per batch
    const int groups = (tiles + UNROLL - 1) / UNROLL;
    const int P = B * BLOCKS_PER_BATCH;

    dim3 grid1(B, BLOCKS_PER_BATCH);
    myLoss_stage1<<<grid1, 256, 0, stream>>>(out_pc, gt_pc, parts, N, groups);
    myLoss_stage2<<<1, 256, 0, stream>>>(parts, out, P, 1.0 / (double)B);
}